// Unitary_33036888441325
// MI455X (gfx1250) — compile-verified
//
#include <hip/hip_runtime.h>

typedef float v2f __attribute__((ext_vector_type(2)));
typedef float v4f __attribute__((ext_vector_type(4)));
typedef float v8f __attribute__((ext_vector_type(8)));

#define REST_N   (1u << 24)          // 2^24 rest columns
#define TILES    (REST_N / 64u)      // 64 rest columns per wave-tile -> 262144 tiles
#define BIT12    (1u << 12)          // target wire 13 stride (elements)
#define BIT20    (1u << 20)          // target wire 5 stride  (elements)

// out[t] = sum_{t'} U[t][t'] * x[t'] over 2^24 rest columns, done as
// D(16x16) = A(16x4) x B(4x16) with A = U padded, B = 16 columns of state data.
__global__ __launch_bounds__(256) void unitary_gate_wmma(
    const float* __restrict__ x,
    const float* __restrict__ U,
    float* __restrict__ out,
    unsigned num_waves)
{
    const unsigned tid  = blockIdx.x * blockDim.x + threadIdx.x;
    const unsigned wave = tid >> 5;
    const unsigned lane = threadIdx.x & 31u;
    const unsigned m    = lane & 15u;
    const unsigned kb   = (lane >> 4) << 1;   // K base: 0 for lanes 0-15, 2 for lanes 16-31

    // A-matrix (16x4 f32, 2 VGPRs): A[m][k] = U[m][k] for m<4, else 0.
    // Documented layout: lane m holds row m; VGPR0 = K=kb, VGPR1 = K=kb+1.
    float a0 = 0.0f, a1 = 0.0f;
    if (m < 4u) {
        a0 = U[m * 4u + kb];
        a1 = U[m * 4u + kb + 1u];
    }
    v2f A;
    A[0] = a0;
    A[1] = a1;

    // Lane 0-15 -> K=0/1 half (base region); lane 16-31 -> K=2/3 half (+2^20).
    const unsigned cLocal = (lane & 15u) * 4u;              // 4 rest cols per lane
    const unsigned hiSel  = (lane & 16u) ? BIT20 : 0u;

    const v8f cz = {};  // zero accumulator

    for (unsigned tile = wave; tile < TILES; tile += num_waves) {
        // rest-index -> flat base: insert zeros at flat bits 12 and 20
        const unsigned r0   = tile << 6;                                  // 64 cols/tile
        const unsigned base = (r0 & 0xFFFu)
                            | (((r0 >> 12) & 0x7Fu) << 13)
                            | ((r0 >> 19) << 21);

        // Two b128 loads cover all 4 K rows for 64 rest columns:
        //   q0: K=0 (lanes 0-15) / K=2 (lanes 16-31)
        //   q1: K=1 (lanes 0-15) / K=3 (lanes 16-31)
        const float* p0 = x + base + hiSel + cLocal;
        const v4f q0 = *(const v4f*)(p0);
        const v4f q1 = *(const v4f*)(p0 + BIT12);

        // B for WMMA j is simply component j of (q0,q1): zero cross-lane shuffles.
        v2f b0; b0[0] = q0[0]; b0[1] = q1[0];
        v2f b1; b1[0] = q0[1]; b1[1] = q1[1];
        v2f b2; b2[0] = q0[2]; b2[1] = q1[2];
        v2f b3; b3[0] = q0[3]; b3[1] = q1[3];

        // args: (neg_a, A, neg_b, B, c_mod, C, reuse_a, reuse_b)
        v8f d0 = __builtin_amdgcn_wmma_f32_16x16x4_f32(false, A, false, b0, (short)0, cz, false, false);
        v8f d1 = __builtin_amdgcn_wmma_f32_16x16x4_f32(false, A, false, b1, (short)0, cz, false, false);
        v8f d2 = __builtin_amdgcn_wmma_f32_16x16x4_f32(false, A, false, b2, (short)0, cz, false, false);
        v8f d3 = __builtin_amdgcn_wmma_f32_16x16x4_f32(false, A, false, b3, (short)0, cz, false, false);

        // D[M=r][N=lane] lives in VGPR r of lanes 0-15; rows 0-3 are out_t.
        // Gather component r across the 4 WMMAs -> contiguous float4 per lane,
        // 256B contiguous per store across lanes 0-15.
        if (lane < 16u) {
            float* o = out + base + cLocal;
            v4f s0 = { d0[0], d1[0], d2[0], d3[0] };   // t=0 -> base
            v4f s1 = { d0[1], d1[1], d2[1], d3[1] };   // t=1 -> base + 2^12
            v4f s2 = { d0[2], d1[2], d2[2], d3[2] };   // t=2 -> base + 2^20
            v4f s3 = { d0[3], d1[3], d2[3], d3[3] };   // t=3 -> base + 2^20 + 2^12
            *(v4f*)(o)                 = s0;
            *(v4f*)(o + BIT12)         = s1;
            *(v4f*)(o + BIT20)         = s2;
            *(v4f*)(o + BIT20 + BIT12) = s3;
        }
    }
}

extern "C" void kernel_launch(void* const* d_in, const int* in_sizes, int n_in,
                              void* d_out, int out_size, void* d_ws, size_t ws_size,
                              hipStream_t stream) {
    (void)in_sizes; (void)n_in; (void)out_size; (void)d_ws; (void)ws_size;
    const float* x = (const float*)d_in[0];   // state vector, 2^26 f32
    const float* U = (const float*)d_in[1];   // 4x4 gate, row-major f32
    float* out = (float*)d_out;               // 2^26 f32

    const unsigned threads = 256;             // 8 wave32 waves per block
    const unsigned blocks  = 4096;            // 32768 waves -> 8 tiles per wave
    const unsigned num_waves = blocks * threads / 32u;

    unitary_gate_wmma<<<dim3(blocks), dim3(threads), 0, stream>>>(x, U, out, num_waves);
}